// ModularLidarGCN_33363305955832
// MI455X (gfx1250) — compile-verified
//
#include <hip/hip_runtime.h>
#include <hip/hip_bf16.h>

typedef __attribute__((ext_vector_type(2))) float v2f;
typedef __attribute__((ext_vector_type(8))) float v8f;

#define BS 256

// ---------------- degree / norm ----------------
__global__ void k_fill(float* __restrict__ p, float v, int n) {
    int i = blockIdx.x * blockDim.x + threadIdx.x;
    if (i < n) p[i] = v;
}

__global__ void k_deg(const int* __restrict__ dst, float* __restrict__ deg, int E) {
    int e = blockIdx.x * blockDim.x + threadIdx.x;
    if (e < E) atomicAdd(&deg[dst[e]], 1.0f);
}

__global__ void k_rsqrt(float* __restrict__ d, int n) {
    int i = blockIdx.x * blockDim.x + threadIdx.x;
    if (i < n) d[i] = __frsqrt_rn(d[i]);
}

// pad x [n,3] -> xp [n,4] (zero 4th column) so layer-0 GEMM is guard-free K=4
__global__ void k_pad3to4(const float* __restrict__ x, float* __restrict__ xp, int n) {
    int idx = blockIdx.x * blockDim.x + threadIdx.x;
    if (idx >= (n << 2)) return;
    int i = idx >> 2, f = idx & 3;
    xp[idx] = (f < 3) ? x[i * 3 + f] : 0.0f;
}

// ---------------- dense GEMM via V_WMMA_F32_16X16X4_F32 ----------------
// H[n, FO] = scale * X[n, K] @ W[K, FO].  K % 4 == 0, n % 16 == 0.
// One wave computes NT adjacent 16x16 output tiles, sharing the A fragment per
// K-step (1 A-load + NT B-loads -> NT WMMAs into independent accumulators).
// No divergence inside active waves: EXEC stays all-ones around every WMMA.
template <int K, int FO, int NT>
__global__ void k_gemm_wmma(const float* __restrict__ X, const float* __restrict__ W,
                            float* __restrict__ H, int n, float scale) {
    constexpr int TN = FO / 16;   // column tiles total
    constexpr int CG = TN / NT;   // column groups (per wave)
    int wave = (blockIdx.x * blockDim.x + threadIdx.x) >> 5;
    int lane = threadIdx.x & 31;
    int tiles = (n >> 4) * CG;
    if (wave >= tiles) return;    // wave-uniform exit
    int tm = wave / CG;
    int cg = wave % CG;

    int m     = lane & 15;            // A row / B column within tile
    int half2 = (lane >> 4) << 1;     // K sub-offset: 0 (lanes 0-15) or 2 (lanes 16-31)
    const float* __restrict__ xr = X + (size_t)((tm << 4) + m) * K + half2;
    int col0 = cg * (NT * 16) + m;

    v8f acc[NT];
#pragma unroll
    for (int t = 0; t < NT; ++t) acc[t] = (v8f){};

#pragma unroll 4
    for (int k = 0; k < K; k += 4) {
        v2f a;
        a.x = xr[k]     * scale;      // contiguous 8B per lane
        a.y = xr[k + 1] * scale;
        const float* __restrict__ w0 = W + (size_t)(k + half2) * FO + col0;
#pragma unroll
        for (int t = 0; t < NT; ++t) {
            v2f b;
            b.x = w0[t * 16];
            b.y = w0[t * 16 + FO];
            acc[t] = __builtin_amdgcn_wmma_f32_16x16x4_f32(false, a, false, b,
                                                           (short)0, acc[t], false, false);
        }
    }

    // C/D layout: VGPR r -> M = r + (lane<16 ? 0 : 8), N = lane&15
    int orow = (tm << 4) + ((lane >> 4) << 3);
#pragma unroll
    for (int t = 0; t < NT; ++t)
#pragma unroll
        for (int r = 0; r < 8; ++r)
            H[(size_t)(orow + r) * FO + col0 + t * 16] = acc[t][r];
}

// ---------------- aggregation ----------------
// out = h * dinv^2 (self-loop) + bias
__global__ void k_init_agg(const float* __restrict__ h, const float* __restrict__ dinv,
                           const float* __restrict__ bias, float* __restrict__ out,
                           int n, int fo_log2) {
    int idx = blockIdx.x * blockDim.x + threadIdx.x;
    if (idx >= (n << fo_log2)) return;
    int i = idx >> fo_log2;
    int f = idx & ((1 << fo_log2) - 1);
    float di = dinv[i];
    out[idx] = h[idx] * di * di + bias[f];
}

// out[dst] += h[src] * dinv[src] * dinv[dst]  (edge-parallel float atomics; h fits in L2)
__global__ void k_scatter(const float* __restrict__ h, const float* __restrict__ dinv,
                          const int* __restrict__ src, const int* __restrict__ dstv,
                          float* __restrict__ out, int E, int fo_log2) {
    int idx = blockIdx.x * blockDim.x + threadIdx.x;
    if (idx >= (E << fo_log2)) return;
    int e = idx >> fo_log2;
    int f = idx & ((1 << fo_log2) - 1);
    int s = src[e];
    int d = dstv[e];
    atomicAdd(&out[(d << fo_log2) + f], h[(s << fo_log2) + f] * dinv[s] * dinv[d]);
}

__global__ void k_relu(float* __restrict__ p, int n) {
    int i = blockIdx.x * blockDim.x + threadIdx.x;
    if (i < n) p[i] = fmaxf(p[i], 0.0f);
}

// ---------------- pooling + head ----------------
__global__ void k_count(const int* __restrict__ batch, float* __restrict__ counts, int n) {
    int i = blockIdx.x * blockDim.x + threadIdx.x;
    if (i < n) atomicAdd(&counts[batch[i]], 1.0f);
}

__global__ void k_pool(const float* __restrict__ h, const int* __restrict__ batch,
                       float* __restrict__ pooled, int n) {
    int idx = blockIdx.x * blockDim.x + threadIdx.x;
    if (idx >= (n << 4)) return;
    int i = idx >> 4;
    int f = idx & 15;
    atomicAdd(&pooled[(batch[i] << 4) + f], h[idx]);
}

__global__ void k_head(const float* __restrict__ pooled, const float* __restrict__ counts,
                       const float* __restrict__ fc1w, const float* __restrict__ fc1b,
                       const float* __restrict__ fc2w, const float* __restrict__ fc2b,
                       float* __restrict__ out, int G) {
    int g = blockIdx.x * blockDim.x + threadIdx.x;
    if (g >= G) return;
    float inv = 1.0f / fmaxf(counts[g], 1.0f);
    float p[16];
#pragma unroll
    for (int k = 0; k < 16; ++k) p[k] = pooled[g * 16 + k] * inv;
    float z[8];
#pragma unroll
    for (int j = 0; j < 8; ++j) {
        float a = fc1b[j];
#pragma unroll
        for (int k = 0; k < 16; ++k) a = fmaf(p[k], fc1w[k * 8 + j], a);
        z[j] = fmaxf(a, 0.0f);
    }
#pragma unroll
    for (int o = 0; o < 4; ++o) {
        float a = fc2b[o];
#pragma unroll
        for (int j = 0; j < 8; ++j) a = fmaf(z[j], fc2w[j * 4 + o], a);
        out[g * 4 + o] = a;
    }
}

static inline int cdiv(long a, int b) { return (int)((a + b - 1) / b); }

extern "C" void kernel_launch(void* const* d_in, const int* in_sizes, int n_in,
                              void* d_out, int out_size, void* d_ws, size_t ws_size,
                              hipStream_t stream) {
    (void)n_in; (void)ws_size;
    // inputs (setup_inputs order)
    const float* x      = (const float*)d_in[0];    // [N,3]
    const int*   ei     = (const int*)d_in[1];      // [2,E]
    const int*   batch  = (const int*)d_in[2];      // [N]
    const float* W[4]   = {(const float*)d_in[3], (const float*)d_in[5],
                           (const float*)d_in[7], (const float*)d_in[9]};
    const float* B[4]   = {(const float*)d_in[4], (const float*)d_in[6],
                           (const float*)d_in[8], (const float*)d_in[10]};
    const float* fc1w   = (const float*)d_in[11];
    const float* fc1b   = (const float*)d_in[12];
    const float* fc2w   = (const float*)d_in[13];
    const float* fc2b   = (const float*)d_in[14];
    float*       out    = (float*)d_out;

    const int N = in_sizes[0] / 3;
    const int E = in_sizes[1] / 2;
    const int G = out_size / 4;
    const int HID = 128;

    const int* src = ei;        // edge_index[0]
    const int* dst = ei + E;    // edge_index[1]

    // workspace layout (256B aligned)
    char* ws = (char*)d_ws;
    size_t off = 0;
    auto take = [&](size_t bytes) { char* p = ws + off; off = (off + bytes + 255) & ~(size_t)255; return p; };
    float* dinv   = (float*)take((size_t)N * sizeof(float));
    float* xpad   = (float*)take((size_t)N * 4 * sizeof(float));
    float* bufA   = (float*)take((size_t)N * HID * sizeof(float));   // GEMM output h
    float* bufB   = (float*)take((size_t)N * HID * sizeof(float));   // aggregated / layer input
    float* pooled = (float*)take((size_t)G * 16 * sizeof(float));
    float* counts = (float*)take((size_t)G * sizeof(float));

    // ---- norms: deg = indegree + 1 (self loop); dinv = rsqrt(deg) ----
    k_fill<<<cdiv(N, BS), BS, 0, stream>>>(dinv, 1.0f, N);
    k_deg<<<cdiv(E, BS), BS, 0, stream>>>(dst, dinv, E);
    k_rsqrt<<<cdiv(N, BS), BS, 0, stream>>>(dinv, N);

    // ---- pad x to K=4 so layer-0 GEMM is guard-free ----
    k_pad3to4<<<cdiv((long)N << 2, BS), BS, 0, stream>>>(x, xpad, N);

    // ---- 4 GCN layers; layer input lives in bufB (layer 0 reads xpad) ----
    const int folog[4] = {7, 6, 5, 4};
    auto gemm_blocks = [&](int fo, int nt) {
        long waves = (long)(N >> 4) * (fo / 16 / nt);
        return cdiv(waves * 32, BS);
    };

    for (int L = 0; L < 4; ++L) {
        // GEMM: bufA = scale * Xin @ W[L]
        switch (L) {
        case 0:
            k_gemm_wmma<4, 128, 4><<<gemm_blocks(128, 4), BS, 0, stream>>>(xpad, W[0], bufA, N, 1.0f);
            break;
        case 1:
            k_gemm_wmma<128, 64, 4><<<gemm_blocks(64, 4), BS, 0, stream>>>(bufB, W[1], bufA, N, 2.0f);
            break;
        case 2:
            k_gemm_wmma<64, 32, 2><<<gemm_blocks(32, 2), BS, 0, stream>>>(bufB, W[2], bufA, N, 2.0f);
            break;
        default:
            k_gemm_wmma<32, 16, 1><<<gemm_blocks(16, 1), BS, 0, stream>>>(bufB, W[3], bufA, N, 2.0f);
            break;
        }

        long nf = (long)N << folog[L];
        k_init_agg<<<cdiv(nf, BS), BS, 0, stream>>>(bufA, dinv, B[L], bufB, N, folog[L]);

        long ef = (long)E << folog[L];
        k_scatter<<<cdiv(ef, BS), BS, 0, stream>>>(bufA, dinv, src, dst, bufB, E, folog[L]);

        k_relu<<<cdiv(nf, BS), BS, 0, stream>>>(bufB, (int)nf);
    }

    // ---- global mean pool + MLP head ----
    k_fill<<<cdiv(G * 16, BS), BS, 0, stream>>>(pooled, 0.0f, G * 16);
    k_fill<<<cdiv(G, BS), BS, 0, stream>>>(counts, 0.0f, G);
    k_count<<<cdiv(N, BS), BS, 0, stream>>>(batch, counts, N);
    k_pool<<<cdiv((long)N << 4, BS), BS, 0, stream>>>(bufB, batch, pooled, N);
    k_head<<<cdiv(G, 64), 64, 0, stream>>>(pooled, counts, fc1w, fc1b, fc2w, fc2b, out, G);
}